// Spiking1DIFLayer_8246337208351
// MI455X (gfx1250) — compile-verified
//
#include <hip/hip_runtime.h>
#include <stdint.h>

typedef uint32_t u32;
typedef uint64_t u64;
typedef u32 v4u __attribute__((ext_vector_type(4)));
typedef u32 v8u __attribute__((ext_vector_type(8)));

#define B_    32
#define C_    512
#define T_    1024
#define TC    128              // timesteps per chunk
#define KCH   (T_ / TC)        // 8 chunks
#define ROWS  32               // neurons (rows) per tile == lanes per wave
#define ROWPAD (TC + 1)        // LDS row stride in dwords (TDM pad: +1 dword per 128)

// Issue one TDM 2D tile load: 32 rows x TC floats, row stride T_ floats in
// global memory, padded to ROWPAD dwords per row in LDS (bank-conflict-free).
// Descriptor built per CDNA5 ISA ch.8 (D# group0 = 4 SGPRs, group1 = 8 SGPRs;
// groups 2/3 omitted => 2D tensor). Tracked with TENSORcnt.
__device__ __forceinline__ void tdm_load_tile(const float* gtile, u32 lds_byte_addr) {
    u64 ga = (u64)(uintptr_t)gtile;
    v4u g0;
    g0[0] = 1u;                                            // count=1 (valid), user mode
    g0[1] = lds_byte_addr;                                 // lds_addr
    g0[2] = (u32)ga;                                       // global_addr[31:0]
    g0[3] = ((u32)(ga >> 32) & 0x01FFFFFFu) | (2u << 30);  // global_addr[56:32] | type=2
    v8u g1;
    g1[0] = (2u << 16)      // data_size = 2 -> 4 bytes
          | (1u << 20)      // pad_enable
          | (6u << 22);     // pad_interval code 6 -> every 128 dwords
                            // pad_amount code 0 -> 1 dword
    g1[1] = ((u32)T_ & 0xFFFFu) << 16;                     // tensor_dim0 = 1024 (lo16)
    g1[2] = ((u32)ROWS & 0xFFFFu) << 16;                   // dim0 hi=0 | tensor_dim1 = 32
    g1[3] = ((u32)TC & 0xFFFFu) << 16;                     // dim1 hi=0 | tile_dim0 = 128
    g1[4] = (u32)ROWS;                                     // tile_dim1 = 32, tile_dim2 = 0
    g1[5] = (u32)T_;                                       // tensor_dim0_stride = 1024 (lo32)
    g1[6] = 0u;                                            // stride0 hi | stride1 lo
    g1[7] = 0u;                                            // stride1 hi
    asm volatile("tensor_load_to_lds %0, %1" :: "s"(g0), "s"(g1) : "memory");
}

__global__ void __launch_bounds__(32)
if_scan_kernel(const float* __restrict__ x, const float* __restrict__ Vth,
               float* __restrict__ out) {
    __shared__ float inbuf[2][ROWS * ROWPAD];   // 2 x 16.1 KB double buffer

    const int lane = threadIdx.x;               // wave32: lane == thread
    const int bid  = blockIdx.x;
    const int b    = bid >> 4;                  // C_/32 = 16 channel groups
    const int c0   = (bid & 15) << 5;           // first channel of this wave
    const size_t tile = ((size_t)b * C_ + c0) * T_;
    const float* xt   = x + tile;               // 32 rows, stride T_
    float* orow       = out + tile + (size_t)lane * T_;
    const float vth   = Vth[c0 + lane];

    // Prefetch chunk 0
    tdm_load_tile(xt, (u32)(uintptr_t)&inbuf[0][0]);

    float mem = 0.0f, spk = 0.0f;
    for (int k = 0; k < KCH; ++k) {
        if (k + 1 < KCH) {
            // Prefetch next chunk into the other buffer, then wait for chunk k
            // (TDM ops complete in order: <=1 outstanding means L_k is done).
            tdm_load_tile(xt + (k + 1) * TC, (u32)(uintptr_t)&inbuf[(k + 1) & 1][0]);
            asm volatile("s_wait_tensorcnt 0x1" ::: "memory");
        } else {
            asm volatile("s_wait_tensorcnt 0x0" ::: "memory");
        }

        const float* row = &inbuf[k & 1][lane * ROWPAD];
        float* och = orow + k * TC;
        #pragma unroll 4
        for (int t = 0; t < TC; t += 4) {
            float4 o; float xv;
            xv = row[t + 0]; mem = mem - spk * vth + xv; spk = (mem > vth) ? 1.0f : 0.0f; o.x = spk;
            xv = row[t + 1]; mem = mem - spk * vth + xv; spk = (mem > vth) ? 1.0f : 0.0f; o.y = spk;
            xv = row[t + 2]; mem = mem - spk * vth + xv; spk = (mem > vth) ? 1.0f : 0.0f; o.z = spk;
            xv = row[t + 3]; mem = mem - spk * vth + xv; spk = (mem > vth) ? 1.0f : 0.0f; o.w = spk;
            *reinterpret_cast<float4*>(och + t) = o;   // 16B store; L2 combines lines
        }
    }
}

extern "C" void kernel_launch(void* const* d_in, const int* in_sizes, int n_in,
                              void* d_out, int out_size, void* d_ws, size_t ws_size,
                              hipStream_t stream) {
    (void)in_sizes; (void)n_in; (void)out_size; (void)d_ws; (void)ws_size;
    const float* x   = (const float*)d_in[0];
    const float* Vth = (const float*)d_in[1];
    float* out       = (float*)d_out;

    dim3 grid(B_ * (C_ / 32));   // 512 wave-sized workgroups
    dim3 block(32);              // one wave32 per workgroup
    if_scan_kernel<<<grid, block, 0, stream>>>(x, Vth, out);
}